// S4DJoint_19550691131559
// MI455X (gfx1250) — compile-verified
//
#include <hip/hip_runtime.h>
#include <math.h>

// Problem constants (match reference)
#define B_   8
#define H_   512
#define L_   2048
#define N2_  32
#define T_   128
#define NC_  (L_ / T_)          // 16 chunks
#define HN_  (H_ * N2_)         // 16384
#define BHL_ ((size_t)B_ * H_ * L_)

typedef float v2f __attribute__((ext_vector_type(2)));
typedef float v8f __attribute__((ext_vector_type(8)));
typedef unsigned int u32x4 __attribute__((ext_vector_type(4)));
typedef int i32x4 __attribute__((ext_vector_type(4)));
typedef int i32x8 __attribute__((ext_vector_type(8)));

__device__ __forceinline__ float gelu_exact(float y) {
    return 0.5f * y * (1.0f + erff(y * 0.70710678118654752f));
}

// ---------------------------------------------------------------------------
// Kernel 1: per-(h,n) coefficients.
// cf layout (12 planes of HN_ floats):
//  0:wr 1:wi 2:wTr 3:wTi 4:2*cm_r 5:2*cm_i 6:2*ca_r 7:2*ca_i
//  8:2*em_r 9:2*em_i 10:2*ea_r 11:2*ea_i
// ---------------------------------------------------------------------------
__global__ void k_coeffs(const float* __restrict__ C, const float* __restrict__ Caux,
                         const float* __restrict__ E, const float* __restrict__ log_dt,
                         const float* __restrict__ log_A_real, const float* __restrict__ A_imag,
                         float* __restrict__ cf) {
    int tid = blockIdx.x * blockDim.x + threadIdx.x;
    if (tid >= HN_) return;
    int h = tid / N2_;
    float dt  = expf(log_dt[h]);
    float Ar  = -expf(log_A_real[tid]);
    float Ai  = A_imag[tid];
    float dar = Ar * dt, dai = Ai * dt;
    float er  = expf(dar);
    float wr  = er * cosf(dai), wi = er * sinf(dai);
    // w^T (chunk-advance factor), T_=128
    float erT = expf(dar * (float)T_);
    float aT  = dai * (float)T_;
    float wTr = erT * cosf(aT), wTi = erT * sinf(aT);
    // scale = (w - 1)/A
    float den = Ar * Ar + Ai * Ai;
    float m1r = wr - 1.0f;
    float sr  = (m1r * Ar + wi * Ai) / den;
    float si  = (wi * Ar - m1r * Ai) / den;
    // C_main = C*scale ; Ca = C_aux*scale ; E_main = E*C_main ; E_aux = E*Ca
    float Cr = C[2 * tid], Ci = C[2 * tid + 1];
    float cmr = Cr * sr - Ci * si, cmi = Cr * si + Ci * sr;
    float Xr = Caux[2 * tid], Xi = Caux[2 * tid + 1];
    float car = Xr * sr - Xi * si, cai = Xr * si + Xi * sr;
    float Er = E[2 * tid], Ei = E[2 * tid + 1];
    float emr = Er * cmr - Ei * cmi, emi = Er * cmi + Ei * cmr;
    float ear = Er * car - Ei * cai, eai = Er * cai + Ei * car;
    cf[0 * HN_ + tid] = wr;          cf[1 * HN_ + tid] = wi;
    cf[2 * HN_ + tid] = wTr;         cf[3 * HN_ + tid] = wTi;
    cf[4 * HN_ + tid] = 2.f * cmr;   cf[5 * HN_ + tid] = 2.f * cmi;
    cf[6 * HN_ + tid] = 2.f * car;   cf[7 * HN_ + tid] = 2.f * cai;
    cf[8 * HN_ + tid] = 2.f * emr;   cf[9 * HN_ + tid] = 2.f * emi;
    cf[10 * HN_ + tid] = 2.f * ear;  cf[11 * HN_ + tid] = 2.f * eai;
}

// ---------------------------------------------------------------------------
// Kernel 2: per-chunk local end-states (zero initial state).
// One wave per (b,h,chunk); lane = state index n.
// ---------------------------------------------------------------------------
__global__ void k_chunk_local(const float* __restrict__ u, const float* __restrict__ x,
                              const float* __restrict__ cf, float4* __restrict__ S) {
    int wid  = blockIdx.x * (blockDim.x >> 5) + (threadIdx.x >> 5);
    int lane = threadIdx.x & 31;
    int c    = wid % NC_;
    int bh   = wid / NC_;
    int h    = bh % H_;
    float wr = cf[0 * HN_ + h * N2_ + lane];
    float wi = cf[1 * HN_ + h * N2_ + lane];
    size_t base = (size_t)bh * L_ + (size_t)c * T_;
    float sur = 0.f, sui = 0.f, sxr = 0.f, sxi = 0.f;
    for (int i = 0; i < T_ / 32; ++i) {
        float uv = u[base + i * 32 + lane];
        float xv = x[base + i * 32 + lane];
        for (int j = 0; j < 32; ++j) {
            float uval = __shfl(uv, j, 32);
            float xval = __shfl(xv, j, 32);
            float nur = wr * sur - wi * sui + uval;
            float nui = wr * sui + wi * sur;
            sur = nur; sui = nui;
            float nxr = wr * sxr - wi * sxi + xval;
            float nxi = wr * sxi + wi * sxr;
            sxr = nxr; sxi = nxi;
        }
    }
    S[(size_t)wid * N2_ + lane] = make_float4(sur, sui, sxr, sxi);
}

// ---------------------------------------------------------------------------
// Kernel 3: sequential scan over chunks -> chunk-START states, in place.
// ---------------------------------------------------------------------------
__global__ void k_chunk_scan(const float* __restrict__ cf, float4* __restrict__ S) {
    int bh   = blockIdx.x * (blockDim.x >> 5) + (threadIdx.x >> 5);
    int lane = threadIdx.x & 31;
    int h    = bh % H_;
    float wTr = cf[2 * HN_ + h * N2_ + lane];
    float wTi = cf[3 * HN_ + h * N2_ + lane];
    float gur = 0.f, gui = 0.f, gxr = 0.f, gxi = 0.f;
    size_t base = (size_t)bh * NC_ * N2_ + lane;
    for (int c = 0; c < NC_; ++c) {
        size_t idx = base + (size_t)c * N2_;
        float4 loc = S[idx];
        S[idx] = make_float4(gur, gui, gxr, gxi);
        float nur = wTr * gur - wTi * gui + loc.x;
        float nui = wTr * gui + wTi * gur + loc.y;
        float nxr = wTr * gxr - wTi * gxi + loc.z;
        float nxi = wTr * gxi + wTi * gxr + loc.w;
        gur = nur; gui = nui; gxr = nxr; gxi = nxi;
    }
}

// ---------------------------------------------------------------------------
// Kernel 4: output pass: recurrence + wave32 butterfly reduce + skip + GELU.
// ---------------------------------------------------------------------------
__global__ void k_ssm_out(const float* __restrict__ u, const float* __restrict__ x,
                          const float* __restrict__ cf, const float4* __restrict__ S,
                          const float* __restrict__ Dm, const float* __restrict__ Fm,
                          const float* __restrict__ Da, const float* __restrict__ Fa,
                          float* __restrict__ Ym, float* __restrict__ Ya) {
    int wid  = blockIdx.x * (blockDim.x >> 5) + (threadIdx.x >> 5);
    int lane = threadIdx.x & 31;
    int c    = wid % NC_;
    int bh   = wid / NC_;
    int h    = bh % H_;
    int cb   = h * N2_ + lane;
    float wr  = cf[0 * HN_ + cb],  wi  = cf[1 * HN_ + cb];
    float cmr = cf[4 * HN_ + cb],  cmi = cf[5 * HN_ + cb];
    float car = cf[6 * HN_ + cb],  cai = cf[7 * HN_ + cb];
    float emr = cf[8 * HN_ + cb],  emi = cf[9 * HN_ + cb];
    float ear = cf[10 * HN_ + cb], eai = cf[11 * HN_ + cb];
    float Dmh = Dm[h], Fmh = Fm[h], Dah = Da[h], Fah = Fa[h];
    float4 g = S[(size_t)wid * N2_ + lane];
    float sur = g.x, sui = g.y, sxr = g.z, sxi = g.w;
    size_t base = (size_t)bh * L_ + (size_t)c * T_;
    for (int i = 0; i < T_ / 32; ++i) {
        float uv = u[base + i * 32 + lane];
        float xv = x[base + i * 32 + lane];
        if (i + 1 < T_ / 32) {   // pull next group toward the WGP (global_prefetch_b8)
            __builtin_prefetch(u + base + (i + 1) * 32 + lane, 0, 1);
            __builtin_prefetch(x + base + (i + 1) * 32 + lane, 0, 1);
        }
        float regm = 0.f, rega = 0.f;
        for (int j = 0; j < 32; ++j) {
            float uval = __shfl(uv, j, 32);
            float xval = __shfl(xv, j, 32);
            float nur = wr * sur - wi * sui + uval;
            float nui = wr * sui + wi * sur;
            sur = nur; sui = nui;
            float nxr = wr * sxr - wi * sxi + xval;
            float nxi = wr * sxi + wi * sxr;
            sxr = nxr; sxi = nxi;
            float pm = cmr * sur - cmi * sui + emr * sxr - emi * sxi;
            float pa = car * sur - cai * sui + ear * sxr - eai * sxi;
            #pragma unroll
            for (int off = 16; off >= 1; off >>= 1) {
                pm += __shfl_xor(pm, off, 32);
                pa += __shfl_xor(pa, off, 32);
            }
            float ym = pm + Dmh * uval + Fmh * xval;
            float ya = pa + Dah * uval + Fah * xval;
            ym = gelu_exact(ym);
            ya = gelu_exact(ya);
            regm = (lane == j) ? ym : regm;
            rega = (lane == j) ? ya : rega;
        }
        Ym[base + i * 32 + lane] = regm;
        Ya[base + i * 32 + lane] = rega;
    }
}

// ---------------------------------------------------------------------------
// Kernel 5: GLU GEMM, fp32 WMMA + TDM-staged A operand.
//
// Block (256 thr = 8 waves) owns one 16-row tile pair of W: a-rows
// [m0,m0+16) and g-rows [m0+H, m0+H+16), all 512 K columns, staged into LDS
// by the Tensor Data Mover.  TDM pad_enable inserts 2 dwords every 256
// dwords -> effective LDS row stride 516 dwords, giving conflict-free
// ds_load_b64 A-fragment reads (bank = (4*row + k) % 64).
// Each wave computes a 64-column slice: 4 N-tiles x (a,g) accumulators,
// 8 v_wmma_f32_16x16x4_f32 per K-step, GLU fused in the epilogue.
// ---------------------------------------------------------------------------
#define LDSW     516            // padded LDS row stride (dwords)
#define WTILE_DW (16 * LDSW)    // 8256 dwords per staged tile

__global__ void k_glu_wmma(const float* __restrict__ Y, const float* __restrict__ W,
                           const float* __restrict__ bias, float* __restrict__ out) {
    extern __shared__ float smem[];            // 2 * WTILE_DW floats (dynamic)
    float* sWa = smem;
    float* sWg = smem + WTILE_DW;

    int cb   = blockIdx.x;
    int ntb  = cb & 3;            // 4 column super-tiles of 512
    int mt   = (cb >> 2) & 31;    // 32 row tiles
    int b    = cb >> 7;           // batch
    int wv   = threadIdx.x >> 5;  // wave in block (0..7)
    int lane = threadIdx.x & 31;
    int m0   = mt * 16;
    int n0   = ntb * 512 + wv * 64;
    int lo   = lane & 15, hi = lane >> 4;
    const float* Yb = Y + (size_t)b * H_ * L_;

    // ---- Tensor Data Mover: stage the two 16x512 f32 W tiles into LDS ----
    if (wv == 0) {
        unsigned long long ga = (unsigned long long)(uintptr_t)(W + (size_t)m0 * H_);
        unsigned long long gg = (unsigned long long)(uintptr_t)(W + (size_t)(m0 + H_) * H_);
        unsigned lds_a = (unsigned)(uintptr_t)sWa;   // low 32 bits of flat = LDS offset
        unsigned lds_g = (unsigned)(uintptr_t)sWg;
        // D# group0: count=1 | lds_addr | global_addr[56:0] | type=2
        u32x4 g0a = { 1u, lds_a, (unsigned)(ga & 0xFFFFFFFFu),
                      (unsigned)((ga >> 32) & 0x01FFFFFFu) | (2u << 30) };
        u32x4 g0g = { 1u, lds_g, (unsigned)(gg & 0xFFFFFFFFu),
                      (unsigned)((gg >> 32) & 0x01FFFFFFu) | (2u << 30) };
        // D# group1: data_size=4B(2) | pad_en | pad_interval=256DW(7) | pad_amount=2DW(1)
        //            tensor 512x16, tile 512x16, dim0 stride 512
        i32x8 g1;
        g1[0] = (2 << 16) | (1 << 20) | (7 << 22) | (1 << 25);
        g1[1] = (int)(512u << 16);   // tensor_dim0[15:0] in [31:16]
        g1[2] = (int)(16u << 16);    // tensor_dim0[31:16]=0 | tensor_dim1[15:0]=16
        g1[3] = (int)(512u << 16);   // tensor_dim1[31:16]=0 | tile_dim0=512
        g1[4] = 16;                  // tile_dim1=16 | tile_dim2=0
        g1[5] = 512;                 // tensor_dim0_stride[31:0]
        g1[6] = 0;
        g1[7] = 0;
        i32x4 z4 = { 0, 0, 0, 0 };                       // groups 2/3 unused (2-D)
        i32x8 z8 = { 0, 0, 0, 0, 0, 0, 0, 0 };           // extra group (unused)
        __builtin_amdgcn_tensor_load_to_lds(g0a, g1, z4, z4, z8, 0);
        __builtin_amdgcn_tensor_load_to_lds(g0g, g1, z4, z4, z8, 0);
        __builtin_amdgcn_s_wait_tensorcnt(0);
    }
    __syncthreads();

    // ---- bias-initialized accumulators ----
    v8f cA, cG;
    #pragma unroll
    for (int r = 0; r < 8; ++r) {
        cA[r] = bias[m0 + r + 8 * hi];
        cG[r] = bias[m0 + H_ + r + 8 * hi];
    }
    v8f accA[4], accG[4];
    #pragma unroll
    for (int j = 0; j < 4; ++j) { accA[j] = cA; accG[j] = cG; }

    // ---- K loop: A frags from LDS (padded layout), B frags from global ----
    for (int k = 0; k < H_; k += 4) {
        int ku  = k + 2 * hi;                    // this half-wave's K pair
        int idx = lo * LDSW + ku + 2 * (ku >> 8);
        float2 ta = *(const float2*)&sWa[idx];
        float2 tg = *(const float2*)&sWg[idx];
        v2f a_a, a_g;
        a_a[0] = ta.x; a_a[1] = ta.y;
        a_g[0] = tg.x; a_g[1] = tg.y;
        #pragma unroll
        for (int j = 0; j < 4; ++j) {
            int col = n0 + j * 16 + lo;
            v2f bf;
            bf[0] = Yb[(size_t)ku * L_ + col];
            bf[1] = Yb[(size_t)(ku + 1) * L_ + col];
            accA[j] = __builtin_amdgcn_wmma_f32_16x16x4_f32(
                false, a_a, false, bf, (short)0, accA[j], false, false);
            accG[j] = __builtin_amdgcn_wmma_f32_16x16x4_f32(
                false, a_g, false, bf, (short)0, accG[j], false, false);
        }
    }

    // ---- Epilogue: GLU and store ----
    #pragma unroll
    for (int j = 0; j < 4; ++j) {
        #pragma unroll
        for (int r = 0; r < 8; ++r) {
            float a  = accA[j][r];
            float gg = accG[j][r];
            float o  = a / (1.0f + expf(-gg));
            int m    = m0 + r + 8 * hi;
            int col  = n0 + j * 16 + lo;
            out[((size_t)b * H_ + m) * L_ + col] = o;
        }
    }
}

// ---------------------------------------------------------------------------
extern "C" void kernel_launch(void* const* d_in, const int* in_sizes, int n_in,
                              void* d_out, int out_size, void* d_ws, size_t ws_size,
                              hipStream_t stream) {
    (void)in_sizes; (void)n_in; (void)out_size; (void)ws_size;
    const float* u          = (const float*)d_in[0];
    const float* x          = (const float*)d_in[1];
    const float* C          = (const float*)d_in[2];
    const float* Caux       = (const float*)d_in[3];
    const float* E          = (const float*)d_in[4];
    const float* log_dt     = (const float*)d_in[5];
    const float* log_A_real = (const float*)d_in[6];
    const float* A_imag     = (const float*)d_in[7];
    const float* Dm         = (const float*)d_in[8];
    const float* Fm         = (const float*)d_in[9];
    const float* Da         = (const float*)d_in[10];
    const float* Fa         = (const float*)d_in[11];
    const float* Wm         = (const float*)d_in[12];
    const float* bm         = (const float*)d_in[13];
    const float* Wx         = (const float*)d_in[14];
    const float* bx         = (const float*)d_in[15];

    // Workspace layout (floats): cf(12*HN) | S(B*H*NC*N2 float4) | Ym | Ya
    float*  ws = (float*)d_ws;
    float*  cf = ws;
    float4* S  = (float4*)(ws + (size_t)12 * HN_);           // 16B aligned
    float*  Ym = ws + (size_t)12 * HN_ + (size_t)4 * B_ * H_ * NC_ * N2_;
    float*  Ya = Ym + BHL_;
    float*  outm = (float*)d_out;
    float*  outa = outm + BHL_;

    k_coeffs<<<HN_ / 256, 256, 0, stream>>>(C, Caux, E, log_dt, log_A_real, A_imag, cf);

    int waves_bc = B_ * H_ * NC_;                            // 65536
    k_chunk_local<<<waves_bc / 8, 256, 0, stream>>>(u, x, cf, S);
    k_chunk_scan<<<(B_ * H_) / 8, 256, 0, stream>>>(cf, S);
    k_ssm_out<<<waves_bc / 8, 256, 0, stream>>>(u, x, cf, S, Dm, Fm, Da, Fa, Ym, Ya);

    int blocks_g = B_ * 32 * 4;                              // 1024 blocks x 8 waves
    size_t lds_bytes = (size_t)2 * WTILE_DW * sizeof(float); // 66 KB dynamic LDS
    k_glu_wmma<<<blocks_g, 256, lds_bytes, stream>>>(Ym, Wm, bm, outm);
    k_glu_wmma<<<blocks_g, 256, lds_bytes, stream>>>(Ya, Wx, bx, outa);
}